// TtMambaBlock_85134841741868
// MI455X (gfx1250) — compile-verified
//
#include <hip/hip_runtime.h>
#include <cmath>

// Mamba decode step, B=32, D=2560, DI=5120, N=16, DT=160, K=4 (conv)
#define BB  32
#define DD  2560
#define DIM 5120
#define NSTATE 16
#define DTR 160
#define XPC (DTR + 2 * NSTATE)   // 192 cols of xdb

typedef __attribute__((ext_vector_type(2))) float v2f;
typedef __attribute__((ext_vector_type(8))) float v8f;

__device__ __forceinline__ float siluf(float v)     { return v / (1.f + expf(-v)); }
__device__ __forceinline__ float softplusf(float v) { return (v > 20.f) ? v : log1pf(expf(v)); }

// ---------------------------------------------------------------------------
// Wave-level GEMM: each wave computes a 32x16 tile of  Out = act(A(32xK) * W(KxN))
// with V_WMMA_F32_16X16X4_F32 (two 16-row M tiles per wave).
// All strides/trip-counts are template constants so the K-loop is a uniform
// scalar loop and every load uses an immediate offset (no per-iter addr math).
// Block = 4 N-tiles (64 cols) x KSPLIT K-splits; splits reduced via LDS.
// ACT: 0 = none, 1 = silu, 2 = softplus(x + bias[col])
// ---------------------------------------------------------------------------
template<int LDA, int LDB, int LDC, int KTOT, int KSPLIT, int ACT>
__global__ __launch_bounds__(32 * 4 * KSPLIT)
void gemm32_wmma(const float* __restrict__ A,
                 const float* __restrict__ W,
                 const float* __restrict__ bias,
                 float* __restrict__ Out)
{
  constexpr int NT   = 4;
  constexpr int KPER = KTOT / KSPLIT;
  static_assert(KPER % 16 == 0, "K chunking requires KPER % 16 == 0");

  const int wave  = (int)threadIdx.x >> 5;
  const int lane  = (int)threadIdx.x & 31;
  const int ntile = wave & (NT - 1);
  const int kspl  = wave >> 2;
  const int n0    = (int)blockIdx.x * (NT * 16) + ntile * 16;
  const int half  = lane >> 4;            // lanes 16-31 carry K+2/K+3 of fragments
  const int l     = lane & 15;
  const int kbase = kspl * KPER + half * 2;

  v8f c0 = {}; v8f c1 = {};
  const float* a0p = A + (size_t)l * LDA + kbase;          // rows 0..15
  const float* a1p = A + (size_t)(l + 16) * LDA + kbase;   // rows 16..31
  const float* bp  = W + (size_t)kbase * LDB + n0 + l;

  for (int k = 0; k < KPER; k += 16) {     // uniform trip count (scalar branch)
    __builtin_prefetch(bp + (size_t)16 * LDB, 0, 0);       // next B chunk
    v2f a0[4], a1[4], b[4];
    #pragma unroll
    for (int j = 0; j < 4; ++j) {          // 16 loads in flight before compute
      a0[j][0] = a0p[4 * j];  a0[j][1] = a0p[4 * j + 1];   // b64, imm offset
      a1[j][0] = a1p[4 * j];  a1[j][1] = a1p[4 * j + 1];
      b[j][0]  = bp[(size_t)(4 * j) * LDB];                // b32, imm offset
      b[j][1]  = bp[(size_t)(4 * j + 1) * LDB];
    }
    #pragma unroll
    for (int j = 0; j < 4; ++j) {
      c0 = __builtin_amdgcn_wmma_f32_16x16x4_f32(false, a0[j], false, b[j], (short)0, c0, false, false);
      c1 = __builtin_amdgcn_wmma_f32_16x16x4_f32(false, a1[j], false, b[j], (short)0, c1, false, false);
    }
    a0p += 16; a1p += 16; bp += (size_t)16 * LDB;
  }

  if constexpr (KSPLIT > 1) {
    __shared__ float red[(KSPLIT - 1) * NT * 512];
    if (kspl > 0) {
      float* p = red + ((size_t)((kspl - 1) * NT + ntile)) * 512 + lane;
      #pragma unroll
      for (int i = 0; i < 8; ++i) { p[i * 32] = c0[i]; p[(i + 8) * 32] = c1[i]; }
    }
    __syncthreads();
    if (kspl == 0) {
      for (int s = 1; s < KSPLIT; ++s) {
        const float* p = red + ((size_t)((s - 1) * NT + ntile)) * 512 + lane;
        #pragma unroll
        for (int i = 0; i < 8; ++i) { c0[i] += p[i * 32]; c1[i] += p[(i + 8) * 32]; }
      }
    }
  }

  if (kspl == 0) {
    const int col = n0 + l;
    float bv = 0.f;
    if (ACT == 2) bv = bias[col];
    #pragma unroll
    for (int i = 0; i < 8; ++i) {
      // C layout: VGPR i holds M=i (lanes 0-15) / M=i+8 (lanes 16-31)
      const int r0 = i + half * 8;
      const int r1 = 16 + i + half * 8;
      float v0 = c0[i], v1 = c1[i];
      if (ACT == 1) { v0 = siluf(v0); v1 = siluf(v1); }
      else if (ACT == 2) { v0 = softplusf(v0 + bv); v1 = softplusf(v1 + bv); }
      Out[(size_t)r0 * LDC + col] = v0;
      Out[(size_t)r1 * LDC + col] = v1;
    }
  }
}

// ---------------------------------------------------------------------------
// Depthwise conv (4-tap, state shifted by one) + bias + silu:
// u[b,d] = silu( cs[1..3][b,d]*cw[0..2][d] + x_ssm[b,d]*cw[3][d] + cb[d] )
// ---------------------------------------------------------------------------
__global__ void conv_silu_kernel(const float* __restrict__ cs,
                                 const float* __restrict__ cw,
                                 const float* __restrict__ cb,
                                 const float* __restrict__ x_ssm,
                                 float* __restrict__ u)
{
  const int idx = (int)blockIdx.x * (int)blockDim.x + (int)threadIdx.x;
  if (idx >= BB * DIM) return;
  const int d = idx % DIM;
  float acc = cb[d];
  acc += cs[(size_t)1 * BB * DIM + idx] * cw[0 * DIM + d];
  acc += cs[(size_t)2 * BB * DIM + idx] * cw[1 * DIM + d];
  acc += cs[(size_t)3 * BB * DIM + idx] * cw[2 * DIM + d];
  acc += x_ssm[idx] * cw[3 * DIM + d];
  u[idx] = siluf(acc);
}

// ---------------------------------------------------------------------------
// Selective-state update + output gating:
// dA = exp(dt * -exp(A_log)); h = ssm_state*dA + dt*u*B; y = h.C + D*u
// z = y * residual      (z is then fed to the out_proj GEMM)
// ---------------------------------------------------------------------------
__global__ void ssm_update_kernel(const float* __restrict__ ssm_state,
                                  const float* __restrict__ A_log,
                                  const float* __restrict__ xdb,
                                  const float* __restrict__ dt,
                                  const float* __restrict__ u,
                                  const float* __restrict__ D_skip,
                                  const float* __restrict__ residual,
                                  float* __restrict__ z)
{
  const int idx = (int)blockIdx.x * (int)blockDim.x + (int)threadIdx.x;
  if (idx >= BB * DIM) return;
  const int b = idx / DIM;
  const int d = idx % DIM;
  const float dtv = dt[idx];
  const float uv  = u[idx];
  const float du  = dtv * uv;
  const float* __restrict__ Brow = xdb + (size_t)b * XPC + DTR;            // cols 160..175
  const float* __restrict__ Crow = xdb + (size_t)b * XPC + DTR + NSTATE;   // cols 176..191
  const float* __restrict__ st   = ssm_state + (size_t)idx * NSTATE;
  const float* __restrict__ al   = A_log + (size_t)d * NSTATE;
  float y = 0.f;
  #pragma unroll
  for (int n = 0; n < NSTATE; ++n) {
    const float dA = expf(-dtv * expf(al[n]));
    const float h  = st[n] * dA + du * Brow[n];
    y += h * Crow[n];
  }
  y += D_skip[d] * uv;
  z[idx] = y * residual[idx];
}

// ---------------------------------------------------------------------------
extern "C" void kernel_launch(void* const* d_in, const int* in_sizes, int n_in,
                              void* d_out, int out_size, void* d_ws, size_t ws_size,
                              hipStream_t stream)
{
  const float* x          = (const float*)d_in[0];   // (1,1,32,2560)
  const float* conv_state = (const float*)d_in[1];   // (4,32,5120)
  const float* ssm_state  = (const float*)d_in[2];   // (32,5120,16)
  const float* ssm_w      = (const float*)d_in[3];   // (2560,5120)
  const float* mlp_w      = (const float*)d_in[4];   // (2560,5120)
  const float* conv_w     = (const float*)d_in[5];   // (4,5120)
  const float* conv_b     = (const float*)d_in[6];   // (5120)
  const float* x_proj_w   = (const float*)d_in[7];   // (5120,192)
  const float* dt_proj_w  = (const float*)d_in[8];   // (160,5120)
  const float* dt_proj_b  = (const float*)d_in[9];   // (5120)
  const float* A_log      = (const float*)d_in[10];  // (5120,16)
  const float* D_skip     = (const float*)d_in[11];  // (5120)
  const float* out_proj_w = (const float*)d_in[12];  // (5120,2560)

  float* ws       = (float*)d_ws;
  const size_t BD = (size_t)BB * DIM;                // 163840
  float* residual = ws + 0 * BD;
  float* x_ssm    = ws + 1 * BD;
  float* u        = ws + 2 * BD;
  float* dt       = ws + 3 * BD;
  float* z        = ws + 4 * BD;
  float* xdb      = ws + 5 * BD;                     // 32*192

  // 1) residual = silu(x @ mlp_w)   [32x2560 * 2560x5120], 52 MB stream
  gemm32_wmma<DD, DIM, DIM, DD, 2, 1><<<dim3(DIM / 64), dim3(256), 0, stream>>>(
      x, mlp_w, nullptr, residual);
  // 2) x_ssm = x @ ssm_w            [32x2560 * 2560x5120], 52 MB stream
  gemm32_wmma<DD, DIM, DIM, DD, 2, 0><<<dim3(DIM / 64), dim3(256), 0, stream>>>(
      x, ssm_w, nullptr, x_ssm);
  // 3) u = silu(conv(states) + b)
  conv_silu_kernel<<<dim3((BB * DIM + 255) / 256), dim3(256), 0, stream>>>(
      conv_state, conv_w, conv_b, x_ssm, u);
  // 4) xdb = u @ x_proj_w           [32x5120 * 5120x192]
  gemm32_wmma<DIM, XPC, XPC, DIM, 4, 0><<<dim3(XPC / 64), dim3(512), 0, stream>>>(
      u, x_proj_w, nullptr, xdb);
  // 5) dt = softplus(dt_r @ dt_proj_w + dt_proj_b)   [32x160 * 160x5120]
  gemm32_wmma<XPC, DIM, DIM, DTR, 1, 2><<<dim3(DIM / 64), dim3(128), 0, stream>>>(
      xdb, dt_proj_w, dt_proj_b, dt);
  // 6) SSM state update + D skip + residual gating -> z
  ssm_update_kernel<<<dim3((BB * DIM + 255) / 256), dim3(256), 0, stream>>>(
      ssm_state, A_log, xdb, dt, u, D_skip, residual, z);
  // 7) out = z @ out_proj_w         [32x5120 * 5120x2560], 52 MB stream
  gemm32_wmma<DIM, DD, DD, DIM, 4, 0><<<dim3(DD / 64), dim3(512), 0, stream>>>(
      z, out_proj_w, nullptr, (float*)d_out);
}